// HNO_41223096107483
// MI455X (gfx1250) — compile-verified
//
#include <hip/hip_runtime.h>
#include <hip/hip_bf16.h>

// ---------------------------------------------------------------------------
// ChebConv GNN forward for MI455X (gfx1250, wave32).
// Memory-bound (~9.5 GB traffic vs ~20 GFLOP dense) -> keep f32 end-to-end and
// use V_WMMA_F32_16X16X4_F32 for all dense GEMMs (free under the HBM roof,
// preserves reference numerics). SpMV = wave-per-edge float4 gather +
// global_atomic_add_f32 scatter. H=128 GEMM: 8-wave workgroup per 16-row
// panel, A staged once in padded LDS (8x traffic reduction, conflict-free
// ds_load_b64), one wave per 16-col tile.
// ---------------------------------------------------------------------------

#define NN 50000
#define EE 800000
#define HH 128

typedef float v2f __attribute__((ext_vector_type(2)));
typedef float v8f __attribute__((ext_vector_type(8)));

// ------------------------------ utilities ---------------------------------

__global__ void k_zero(float* __restrict__ p, long n) {
  long i = (long)blockIdx.x * blockDim.x + threadIdx.x;
  if (i < n) p[i] = 0.0f;
}

__global__ void k_setneg(float* __restrict__ out, const float* __restrict__ a, long n) {
  long i = (long)blockIdx.x * blockDim.x + threadIdx.x;
  if (i < n) out[i] = -a[i];
}

// ------------------------- graph normalization ----------------------------

__global__ void k_degree(const int* __restrict__ src, float* __restrict__ deg) {
  int e = blockIdx.x * blockDim.x + threadIdx.x;
  if (e < EE) atomicAdd(&deg[src[e]], 1.0f);
}

__global__ void k_dinv(const float* __restrict__ deg, float* __restrict__ dinv) {
  int i = blockIdx.x * blockDim.x + threadIdx.x;
  if (i < NN) {
    float d = deg[i];
    dinv[i] = (d > 0.0f) ? rsqrtf(fmaxf(d, 1.0f)) : 0.0f;
  }
}

__global__ void k_edgew(const int* __restrict__ src, const int* __restrict__ dst,
                        const float* __restrict__ dinv, float* __restrict__ wE) {
  int e = blockIdx.x * blockDim.x + threadIdx.x;
  if (e < EE) wE[e] = -dinv[src[e]] * dinv[dst[e]];
}

// ------------------------------- SpMV -------------------------------------

// C = 3 (first layer input): one thread per edge.
__global__ void k_spmv3(const float* __restrict__ h, const int* __restrict__ src,
                        const int* __restrict__ dst, const float* __restrict__ wE,
                        float scale, float* __restrict__ out) {
  int e = blockIdx.x * blockDim.x + threadIdx.x;
  if (e >= EE) return;
  int s = src[e], d = dst[e];
  float c = scale * wE[e];
  atomicAdd(&out[(long)d * 3 + 0], c * h[(long)s * 3 + 0]);
  atomicAdd(&out[(long)d * 3 + 1], c * h[(long)s * 3 + 1]);
  atomicAdd(&out[(long)d * 3 + 2], c * h[(long)s * 3 + 2]);
}

// C = 128: one wave (32 lanes) per edge, float4 per lane.
__global__ __launch_bounds__(256) void k_spmv128(
    const float* __restrict__ h, const int* __restrict__ src,
    const int* __restrict__ dst, const float* __restrict__ wE,
    float scale, float* __restrict__ out) {
  int e = blockIdx.x * 8 + (threadIdx.x >> 5);
  if (e >= EE) return;
  int lane = threadIdx.x & 31;
  int s = src[e], d = dst[e];
  float c = scale * wE[e];
  const float4* hp = (const float4*)(h + (long)s * HH);
  float4 v = hp[lane];
  float* op = out + (long)d * HH + lane * 4;
  atomicAdd(op + 0, c * v.x);
  atomicAdd(op + 1, c * v.y);
  atomicAdd(op + 2, c * v.z);
  atomicAdd(op + 3, c * v.w);
}

// ------------------------- WMMA Chebyshev GEMMs ---------------------------
// out = [T0|T1|T2|T3] @ W + bias,  W = [4][C][H] row-major (so [4C][H]).
// ISA fragment layouts (wave32), V_WMMA_F32_16X16X4_F32:
//   A 16x4 : lanes 0-15 row=lane, K={k0,k0+1}; lanes 16-31 row=lane-16, K={k0+2,k0+3}
//   B 4x16 : lanes 0-15 col=lane, K={k0,k0+1}; lanes 16-31 col=lane-16, K={k0+2,k0+3}
//   C/D    : lane&15 = col; vgpr j -> row j (lanes 0-15) / row j+8 (lanes 16-31)

// --- C=3 (K=12 total): one wave per 16x16 tile, 3 WMMA steps. ---
__global__ __launch_bounds__(32) void k_cheb_gemm3(
    const float* __restrict__ T0, const float* __restrict__ T1,
    const float* __restrict__ T2, const float* __restrict__ T3,
    const float* __restrict__ W, const float* __restrict__ bias,
    float* __restrict__ out) {
  const int lane = threadIdx.x;
  const int tm = blockIdx.x;                 // N/16 tiles
  const int tn = blockIdx.y;                 // H/16 tiles
  const int row = tm * 16 + (lane & 15);
  const int col = tn * 16 + (lane & 15);
  const int khalf = (lane >> 4) * 2;
  const float* Ts[4] = {T0, T1, T2, T3};

  v8f acc = {};
  for (int k0 = 0; k0 < 12; k0 += 4) {
    v2f a, b;
#pragma unroll
    for (int j = 0; j < 2; ++j) {
      int kk = k0 + khalf + j;
      int t = kk / 3;
      int cc = kk - t * 3;
      a[j] = Ts[t][(long)row * 3 + cc];
      b[j] = W[(long)kk * HH + col];
    }
    acc = __builtin_amdgcn_wmma_f32_16x16x4_f32(
        false, a, false, b, (short)0, acc, false, false);
  }

  const float bc = bias[col];
#pragma unroll
  for (int j = 0; j < 8; ++j) {
    int r = tm * 16 + j + (lane >> 4) * 8;
    out[(long)r * HH + (tn * 16 + (lane & 15))] = acc[j] + bc;
  }
}

// --- C=128 (K=512 total): 8-wave block per 16-row panel; A staged in LDS. ---
#define KT 512                // 4*C
#define LST (KT + 4)          // pad 4 floats: row stride 516 -> bank-conflict free

__device__ __forceinline__ void stage_panel(const float* __restrict__ A,
                                            float* __restrict__ As,
                                            int tm, int t, int tid) {
  // copy 16 rows x 128 cols of A into As[row*LST + t*128 + c], float4 chunks
  for (int i = tid; i < 16 * (HH / 4); i += 256) {   // 512 float4s, 2 iters
    int rowl = i >> 5;                               // / (128/4)
    int c4 = i & 31;
    float4 v = ((const float4*)(A + (long)(tm * 16 + rowl) * HH))[c4];
    *(float4*)&As[rowl * LST + t * HH + c4 * 4] = v;
  }
}

__global__ __launch_bounds__(256) void k_cheb_gemm128(
    const float* __restrict__ T0, const float* __restrict__ T1,
    const float* __restrict__ T2, const float* __restrict__ T3,
    const float* __restrict__ W, const float* __restrict__ bias,
    float* __restrict__ out) {
  __shared__ float As[16 * LST];                     // ~33 KB (of 320 KB/WGP)
  const int tid = threadIdx.x;
  const int lane = tid & 31;
  const int wave = tid >> 5;                         // col tile 0..7
  const int tm = blockIdx.x;                         // row tile 0..3124

  stage_panel(T0, As, tm, 0, tid);
  stage_panel(T1, As, tm, 1, tid);
  stage_panel(T2, As, tm, 2, tid);
  stage_panel(T3, As, tm, 3, tid);
  __syncthreads();

  const int l16 = lane & 15;
  const int khalf = (lane >> 4) * 2;
  const int col = wave * 16 + l16;

  v8f acc = {};
#pragma unroll 8
  for (int k0 = 0; k0 < KT; k0 += 4) {
    v2f a = *(const v2f*)&As[l16 * LST + k0 + khalf];  // ds_load_b64
    v2f b;
    b[0] = W[(long)(k0 + khalf) * HH + col];
    b[1] = W[(long)(k0 + khalf + 1) * HH + col];
    acc = __builtin_amdgcn_wmma_f32_16x16x4_f32(
        false, a, false, b, (short)0, acc, false, false);
  }

  const float bc = bias[col];
#pragma unroll
  for (int j = 0; j < 8; ++j) {
    int r = tm * 16 + j + (lane >> 4) * 8;
    out[(long)r * HH + col] = acc[j] + bc;
  }
}

// -------------------- activation + BatchNorm (training) -------------------

// Pass 1: apply activation in place, accumulate per-channel sum / sumsq.
__global__ __launch_bounds__(128) void k_act_stats(float* __restrict__ h,
                                                   float* __restrict__ stats,
                                                   int actType) {
  const int ch = threadIdx.x;
  float s = 0.0f, s2 = 0.0f;
  for (int r = blockIdx.x; r < NN; r += gridDim.x) {
    long rn = (long)(r + gridDim.x);
    if (rn < NN) __builtin_prefetch(&h[rn * HH + ch], 0, 1);  // global_prefetch_b8
    float v = h[(long)r * HH + ch];
    v = (actType == 0) ? (v > 0.0f ? v : 0.01f * v)   // leaky_relu (slope 0.01)
                       : fmaxf(v, 0.0f);              // relu
    h[(long)r * HH + ch] = v;
    s += v;
    s2 += v * v;
  }
  atomicAdd(&stats[ch], s);
  atomicAdd(&stats[HH + ch], s2);
}

// Pass 2: h = (h - mean) * rsqrt(var + eps) * g + be  (population var, ddof=0)
__global__ __launch_bounds__(128) void k_bn_apply(float* __restrict__ h,
                                                  const float* __restrict__ stats,
                                                  const float* __restrict__ g,
                                                  const float* __restrict__ be) {
  const int ch = threadIdx.x;
  const float invN = 1.0f / (float)NN;
  float m = stats[ch] * invN;
  float var = stats[HH + ch] * invN - m * m;
  float sc = g[ch] * rsqrtf(var + 1e-5f);
  float sh = be[ch] - m * sc;
  for (int r = blockIdx.x; r < NN; r += gridDim.x) {
    long idx = (long)r * HH + ch;
    h[idx] = h[idx] * sc + sh;
  }
}

// ---------------- final: row-normalize + [N,128]@[128,3] ------------------
__global__ __launch_bounds__(256) void k_final(const float* __restrict__ h,
                                               const float* __restrict__ Wm,
                                               const float* __restrict__ bm,
                                               float* __restrict__ out) {
  int n = blockIdx.x * 8 + (threadIdx.x >> 5);
  if (n >= NN) return;
  int lane = threadIdx.x & 31;
  const float4* hp = (const float4*)(h + (long)n * HH);
  float4 v = hp[lane];
  int c = lane * 4;
  float ns = v.x * v.x + v.y * v.y + v.z * v.z + v.w * v.w;
  float d0 = v.x * Wm[(c + 0) * 3 + 0] + v.y * Wm[(c + 1) * 3 + 0] +
             v.z * Wm[(c + 2) * 3 + 0] + v.w * Wm[(c + 3) * 3 + 0];
  float d1 = v.x * Wm[(c + 0) * 3 + 1] + v.y * Wm[(c + 1) * 3 + 1] +
             v.z * Wm[(c + 2) * 3 + 1] + v.w * Wm[(c + 3) * 3 + 1];
  float d2 = v.x * Wm[(c + 0) * 3 + 2] + v.y * Wm[(c + 1) * 3 + 2] +
             v.z * Wm[(c + 2) * 3 + 2] + v.w * Wm[(c + 3) * 3 + 2];
#pragma unroll
  for (int off = 16; off > 0; off >>= 1) {
    ns += __shfl_xor(ns, off, 32);
    d0 += __shfl_xor(d0, off, 32);
    d1 += __shfl_xor(d1, off, 32);
    d2 += __shfl_xor(d2, off, 32);
  }
  if (lane == 0) {
    float inv = 1.0f / fmaxf(sqrtf(ns), 1e-12f);
    out[(long)n * 3 + 0] = d0 * inv + bm[0];
    out[(long)n * 3 + 1] = d1 * inv + bm[1];
    out[(long)n * 3 + 2] = d2 * inv + bm[2];
  }
}

// ------------------------------ launcher ----------------------------------

static inline int cdiv(long a, long b) { return (int)((a + b - 1) / b); }

extern "C" void kernel_launch(void* const* d_in, const int* in_sizes, int n_in,
                              void* d_out, int out_size, void* d_ws, size_t ws_size,
                              hipStream_t stream) {
  const float* x   = (const float*)d_in[0];
  const int*   ei  = (const int*)d_in[1];
  const float* W1  = (const float*)d_in[2];
  const float* b1  = (const float*)d_in[3];
  const float* g1  = (const float*)d_in[4];
  const float* be1 = (const float*)d_in[5];
  const float* W2  = (const float*)d_in[6];
  const float* b2  = (const float*)d_in[7];
  const float* g2  = (const float*)d_in[8];
  const float* be2 = (const float*)d_in[9];
  const float* W3  = (const float*)d_in[10];
  const float* b3  = (const float*)d_in[11];
  const float* g3  = (const float*)d_in[12];
  const float* be3 = (const float*)d_in[13];
  const float* W4  = (const float*)d_in[14];
  const float* b4  = (const float*)d_in[15];
  const float* Wm  = (const float*)d_in[16];
  const float* bm  = (const float*)d_in[17];
  float* out = (float*)d_out;

  const int* src = ei;        // edge_index[0]
  const int* dst = ei + EE;   // edge_index[1]

  // workspace layout (floats)
  float* ws   = (float*)d_ws;
  float* deg  = ws;                       // N
  float* dinv = deg + NN;                 // N
  float* wE   = dinv + NN;                // E
  float* T1   = wE + EE;                  // N*H
  float* T2   = T1 + (long)NN * HH;       // N*H
  float* T3   = T2 + (long)NN * HH;       // N*H
  float* hA   = T3 + (long)NN * HH;       // N*H
  float* hB   = hA + (long)NN * HH;       // N*H
  float* stats = hB + (long)NN * HH;      // 2*H

  const long NH = (long)NN * HH;
  const dim3 g3grid(NN / 16, HH / 16);    // 3125 x 8, one wave32 per tile
  const int g128grid = NN / 16;           // 3125 blocks x 8 waves

  // --- graph normalization coefficients ---
  k_zero<<<cdiv(NN, 256), 256, 0, stream>>>(deg, NN);
  k_degree<<<cdiv(EE, 256), 256, 0, stream>>>(src, deg);
  k_dinv<<<cdiv(NN, 256), 256, 0, stream>>>(deg, dinv);
  k_edgew<<<cdiv(EE, 256), 256, 0, stream>>>(src, dst, dinv, wE);

  // ================= Layer 1 (C=3, leaky_relu + BN) =================
  k_zero<<<cdiv((long)NN * 3, 256), 256, 0, stream>>>(T1, (long)NN * 3);
  k_spmv3<<<cdiv(EE, 256), 256, 0, stream>>>(x, src, dst, wE, 1.0f, T1);
  k_setneg<<<cdiv((long)NN * 3, 256), 256, 0, stream>>>(T2, x, (long)NN * 3);
  k_spmv3<<<cdiv(EE, 256), 256, 0, stream>>>(T1, src, dst, wE, 2.0f, T2);
  k_setneg<<<cdiv((long)NN * 3, 256), 256, 0, stream>>>(T3, T1, (long)NN * 3);
  k_spmv3<<<cdiv(EE, 256), 256, 0, stream>>>(T2, src, dst, wE, 2.0f, T3);
  k_cheb_gemm3<<<g3grid, 32, 0, stream>>>(x, T1, T2, T3, W1, b1, hA);
  k_zero<<<cdiv(2 * HH, 256), 256, 0, stream>>>(stats, 2 * HH);
  k_act_stats<<<512, 128, 0, stream>>>(hA, stats, 0);
  k_bn_apply<<<512, 128, 0, stream>>>(hA, stats, g1, be1);

  // ================= Layer 2 (C=128, leaky_relu + BN) =================
  k_zero<<<cdiv(NH, 256), 256, 0, stream>>>(T1, NH);
  k_spmv128<<<cdiv(EE, 8), 256, 0, stream>>>(hA, src, dst, wE, 1.0f, T1);
  k_setneg<<<cdiv(NH, 256), 256, 0, stream>>>(T2, hA, NH);
  k_spmv128<<<cdiv(EE, 8), 256, 0, stream>>>(T1, src, dst, wE, 2.0f, T2);
  k_setneg<<<cdiv(NH, 256), 256, 0, stream>>>(T3, T1, NH);
  k_spmv128<<<cdiv(EE, 8), 256, 0, stream>>>(T2, src, dst, wE, 2.0f, T3);
  k_cheb_gemm128<<<g128grid, 256, 0, stream>>>(hA, T1, T2, T3, W2, b2, hB);
  k_zero<<<cdiv(2 * HH, 256), 256, 0, stream>>>(stats, 2 * HH);
  k_act_stats<<<512, 128, 0, stream>>>(hB, stats, 0);
  k_bn_apply<<<512, 128, 0, stream>>>(hB, stats, g2, be2);

  // ================= Layer 3 (C=128, relu + BN) =================
  k_zero<<<cdiv(NH, 256), 256, 0, stream>>>(T1, NH);
  k_spmv128<<<cdiv(EE, 8), 256, 0, stream>>>(hB, src, dst, wE, 1.0f, T1);
  k_setneg<<<cdiv(NH, 256), 256, 0, stream>>>(T2, hB, NH);
  k_spmv128<<<cdiv(EE, 8), 256, 0, stream>>>(T1, src, dst, wE, 2.0f, T2);
  k_setneg<<<cdiv(NH, 256), 256, 0, stream>>>(T3, T1, NH);
  k_spmv128<<<cdiv(EE, 8), 256, 0, stream>>>(T2, src, dst, wE, 2.0f, T3);
  k_cheb_gemm128<<<g128grid, 256, 0, stream>>>(hB, T1, T2, T3, W3, b3, hA);
  k_zero<<<cdiv(2 * HH, 256), 256, 0, stream>>>(stats, 2 * HH);
  k_act_stats<<<512, 128, 0, stream>>>(hA, stats, 1);
  k_bn_apply<<<512, 128, 0, stream>>>(hA, stats, g3, be3);

  // ================= Layer 4 (C=128, no act/BN) =================
  k_zero<<<cdiv(NH, 256), 256, 0, stream>>>(T1, NH);
  k_spmv128<<<cdiv(EE, 8), 256, 0, stream>>>(hA, src, dst, wE, 1.0f, T1);
  k_setneg<<<cdiv(NH, 256), 256, 0, stream>>>(T2, hA, NH);
  k_spmv128<<<cdiv(EE, 8), 256, 0, stream>>>(T1, src, dst, wE, 2.0f, T2);
  k_setneg<<<cdiv(NH, 256), 256, 0, stream>>>(T3, T1, NH);
  k_spmv128<<<cdiv(EE, 8), 256, 0, stream>>>(T2, src, dst, wE, 2.0f, T3);
  k_cheb_gemm128<<<g128grid, 256, 0, stream>>>(hA, T1, T2, T3, W4, b4, hB);

  // ================= normalize + final projection =================
  k_final<<<cdiv(NN, 8), 256, 0, stream>>>(hB, Wm, bm, out);
}